// HW2DModel_6957847020062
// MI455X (gfx1250) — compile-verified
//
#include <hip/hip_runtime.h>
#include <math.h>

#define NXY 2048
#define NN  (NXY * NXY)
#define K0f      0.15f
#define KAPPAf   1.0f
#define ALPHAf   1.0f
#define DNf      0.001f
#define DOMEGAf  0.001f
#define K2MINf   1e-12f

// Kill-switch for the async global->LDS path in the transpose (inline asm).
#ifndef USE_ASYNC_TRANSPOSE
#define USE_ASYNC_TRANSPOSE 1
#endif

typedef __attribute__((ext_vector_type(2))) float v2f;
typedef __attribute__((ext_vector_type(8))) float v8f;

// kx = K0 * fftfreq_index (L = 2*pi/K0 makes 2*pi*fftfreq == K0 * fi)
__device__ __forceinline__ float kfreq(int i) {
    return K0f * (float)(i - ((i >= NXY / 2) ? NXY : 0));
}

enum { ST_PLAIN = 0, ST_MASK = 1, ST_COMBINE = 2 };

// ---------------------------------------------------------------------------
// Core 2048-point complex FFT on one LDS-resident line.
// 2048 = 128 (batched radix-2 Stockham in LDS) x 16 (DFT-16 as a complex
// 16x16x16 GEMM on tensor cores via V_WMMA_F32_16X16X4_F32; the N=2048
// twiddles are fused into the B-fragment load).  dir=-1 fwd, +1 inv.
// All twiddles come from two 128-entry LDS tables of power-of-two roots:
//   Tw128[t] = exp(dir*2pi*i*t/128)   (Stockham: index p*ss < 64;
//                                      W16^j = Tw128[8j] for the DFT-16)
//   TwB[t]   = exp(dir*2pi*i*t/2048)  (W2048^t = TwB[t&127]*Tw128[8*(t>>7)])
// This keeps the TRANS pipe free around the WMMA ops.
// Caller loads Ar/Ai (layout (n&15)*128 + (n>>4)) and has done __syncthreads.
// STMODE selects the store: plain line store, 2/3-dealias-masked store, or
// final combine (dn/dw) store straight to the output array.
// ---------------------------------------------------------------------------
template <int STMODE>
__device__ __forceinline__ void fft_line_core(
    float* Ar, float* Ai, float* Br, float* Bi,
    float dir, float scale, int line,
    float2* __restrict__ dst_line,
    const float2* __restrict__ Laux,   // ST_COMBINE: packed linear terms (natural)
    float* __restrict__ outbuf)        // ST_COMBINE: d_out
{
    __shared__ float2 Tw128[128];
    __shared__ float2 TwB[128];
    const int tid = threadIdx.x;

    if (tid < 128) {
        float s, c;
        __sincosf(dir * 0.04908738521234052f * (float)tid, &s, &c);   // 2pi/128
        Tw128[tid] = make_float2(c, s);
        __sincosf(dir * 0.00306796157577128245f * (float)tid, &s, &c); // 2pi/2048
        TwB[tid] = make_float2(c, s);
    }
    __syncthreads();

    // 7 radix-2 Stockham (autosort) stages on each 128-point block.
    float *sr = Ar, *si = Ai, *dr = Br, *di = Bi;
    int nn = 128, ss = 1;
    #pragma unroll 1
    for (int stage = 0; stage < 7; ++stage) {
        const int m = nn >> 1;
        for (int u = tid; u < 1024; u += 256) {
            const int blk = u >> 6;          // decimation residue n1
            const int v   = u & 63;          // m*ss == 64 always
            const int p   = v / ss;
            const int q   = v - p * ss;
            const int ib  = blk * 128;
            const float2 w = Tw128[p * ss];  // exp(dir*2pi*i*p/nn)
            const int i0 = ib + q + ss * p;
            const int i1 = i0 + ss * m;
            const float ar = sr[i0], ai = si[i0];
            const float br = sr[i1], bi = si[i1];
            const int o0 = ib + q + ss * 2 * p;
            const int o1 = o0 + ss;
            dr[o0] = ar + br;  di[o0] = ai + bi;
            const float tr = ar - br, ti = ai - bi;
            dr[o1] = tr * w.x - ti * w.y;
            di[o1] = tr * w.y + ti * w.x;
        }
        __syncthreads();
        float* t;
        t = sr; sr = dr; dr = t;
        t = si; si = di; di = t;
        nn >>= 1; ss <<= 1;
    }
    // sr/si hold G[n1*128 + k2], k2 natural order.

    const int lane = tid & 31;
    const int wv   = tid >> 5;
    const int g    = lane >> 4;
    const int mrow = lane & 15;
    const int col  = wv * 16 + (lane & 15);

    v8f cr = {0.f,0.f,0.f,0.f,0.f,0.f,0.f,0.f};
    v8f ci = {0.f,0.f,0.f,0.f,0.f,0.f,0.f,0.f};

#if __has_builtin(__builtin_amdgcn_wmma_f32_16x16x4_f32)
    #pragma unroll
    for (int cch = 0; cch < 4; ++cch) {
        const int kk0 = 4 * cch + 2 * g;
        const int kk1 = kk0 + 1;
        // A-fragment: W16[mrow][kk] = Tw128[8 * ((mrow*kk) & 15)]
        const float2 wa0 = Tw128[((mrow * kk0) & 15) * 8];
        const float2 wa1 = Tw128[((mrow * kk1) & 15) * 8];
        const v2f Arf  = { wa0.x,  wa1.x};
        const v2f Aif  = { wa0.y,  wa1.y};
        const v2f nAif = {-wa0.y, -wa1.y};
        // B-fragment: G[kk][col] * W2048^((kk*col) mod 2048)
        const float gr0 = sr[kk0 * 128 + col], gi0 = si[kk0 * 128 + col];
        const float gr1 = sr[kk1 * 128 + col], gi1 = si[kk1 * 128 + col];
        const int t0 = (kk0 * col) & 2047;
        const int t1 = (kk1 * col) & 2047;
        const float2 b0a = TwB[t0 & 127], b0b = Tw128[(t0 >> 7) * 8];
        const float2 b1a = TwB[t1 & 127], b1b = Tw128[(t1 >> 7) * 8];
        const float tc0 = b0a.x * b0b.x - b0a.y * b0b.y;
        const float ts0 = b0a.x * b0b.y + b0a.y * b0b.x;
        const float tc1 = b1a.x * b1b.x - b1a.y * b1b.y;
        const float ts1 = b1a.x * b1b.y + b1a.y * b1b.x;
        const v2f Brf = {gr0 * tc0 - gi0 * ts0, gr1 * tc1 - gi1 * ts1};
        const v2f Bif = {gr0 * ts0 + gi0 * tc0, gr1 * ts1 + gi1 * tc1};
        cr = __builtin_amdgcn_wmma_f32_16x16x4_f32(false, Arf,  false, Brf, (short)0, cr, false, false);
        cr = __builtin_amdgcn_wmma_f32_16x16x4_f32(false, nAif, false, Bif, (short)0, cr, false, false);
        ci = __builtin_amdgcn_wmma_f32_16x16x4_f32(false, Aif,  false, Brf, (short)0, ci, false, false);
        ci = __builtin_amdgcn_wmma_f32_16x16x4_f32(false, Arf,  false, Bif, (short)0, ci, false, false);
    }
#else
    for (int r = 0; r < 8; ++r) {
        const int k1 = r + 8 * g;
        float accr = 0.f, acci = 0.f;
        for (int kk = 0; kk < 16; ++kk) {
            float wi, wr;
            __sincosf(dir * 0.392699081698724155f * (float)((k1 * kk) & 15), &wi, &wr);
            const float grv = sr[kk * 128 + col], giv = si[kk * 128 + col];
            float ts, tc;
            __sincosf(dir * 0.00306796157577128245f * (float)((kk * col) & 2047), &ts, &tc);
            const float brv = grv * tc - giv * ts;
            const float biv = grv * ts + giv * tc;
            accr += wr * brv - wi * biv;
            acci += wr * biv + wi * brv;
        }
        cr[r] = accr; ci[r] = acci;
    }
#endif

    const float lim = (2.0f / 3.0f) * 1024.0f;
    const float fline = fabsf((float)(line - ((line >= 1024) ? 2048 : 0)));
    #pragma unroll
    for (int r = 0; r < 8; ++r) {
        const int k1  = r + 8 * g;
        const int idx = col + 128 * k1;
        float vr = cr[r] * scale, vi = ci[r] * scale;
        if constexpr (STMODE == ST_MASK) {
            const float fi = fabsf((float)(idx - ((idx >= 1024) ? 2048 : 0)));
            const float msk = (fi <= lim && fline <= lim) ? 1.0f : 0.0f;
            dst_line[idx] = make_float2(vr * msk, vi * msk);
        } else if constexpr (STMODE == ST_COMBINE) {
            const int p = line * NXY + idx;
            const float2 L = Laux[p];
            outbuf[p]      = -vr + L.x;   // dn = -adv_n + lin_n
            outbuf[NN + p] = -vi + L.y;   // dw = -adv_w + lin_w
        } else {
            dst_line[idx] = make_float2(vr, vi);
        }
    }
}

#define FFT_SHARED \
    __shared__ float Ar[2048]; __shared__ float Ai[2048]; \
    __shared__ float Br[2048]; __shared__ float Bi[2048];

#define FFT_DEST(u) (((u) & 15) * 128 + ((u) >> 4))

// -------- plain pass: load line, FFT, store line (in place) ----------------
__global__ __launch_bounds__(256) void fft_pass_kernel(
    float2* __restrict__ data, float dir, float scale)
{
    FFT_SHARED
    const int line = blockIdx.x;
    float2* ln = data + (size_t)line * NXY;
    for (int u = threadIdx.x; u < 2048; u += 256) {
        const float2 v = ln[u];
        Ar[FFT_DEST(u)] = v.x; Ai[FFT_DEST(u)] = v.y;
    }
    __syncthreads();
    fft_line_core<ST_PLAIN>(Ar, Ai, Br, Bi, dir, scale, line, ln, nullptr, nullptr);
}

// -------- fwd pass1 of Z with pack fused: z = n + i*omega ------------------
__global__ __launch_bounds__(256) void fft_pack_kernel(
    const float* __restrict__ a, const float* __restrict__ b,
    float2* __restrict__ dst, float dir, float scale)
{
    FFT_SHARED
    const int line = blockIdx.x;
    const size_t base = (size_t)line * NXY;
    for (int u = threadIdx.x; u < 2048; u += 256) {
        Ar[FFT_DEST(u)] = a[base + u];
        Ai[FFT_DEST(u)] = b[base + u];
    }
    __syncthreads();
    fft_line_core<ST_PLAIN>(Ar, Ai, Br, Bi, dir, scale, line,
                            dst + base, nullptr, nullptr);
}

// -------- inv pass1 with packed-gradient spectrum fused --------------------
// Layout [ky][kx]: line = ky index, u = kx index.
// P = (i*kx*F) + i*(i*ky*F);  isPhi: F = -src/k2.
__global__ __launch_bounds__(256) void fft_gradpack_kernel(
    const float2* __restrict__ src, float2* __restrict__ dst, int isPhi,
    float dir, float scale)
{
    FFT_SHARED
    const int line = blockIdx.x;
    const size_t base = (size_t)line * NXY;
    const float ky = kfreq(line);
    for (int u = threadIdx.x; u < 2048; u += 256) {
        const float kx = kfreq(u);
        const float k2v = kx * kx + ky * ky;
        float2 F = src[base + u];
        if (isPhi) {
            const float inv = (k2v > K2MINf) ? (-1.0f / k2v) : 0.0f;
            F = make_float2(inv * F.x, inv * F.y);
        }
        Ar[FFT_DEST(u)] = -kx * F.y - ky * F.x;
        Ai[FFT_DEST(u)] =  kx * F.x - ky * F.y;
    }
    __syncthreads();
    fft_line_core<ST_PLAIN>(Ar, Ai, Br, Bi, dir, scale, line,
                            dst + base, nullptr, nullptr);
}

// -------- inv pass1 with all linear terms fused ----------------------------
// Ln = -kappa*i*ky*phi + alpha*(phi-n) - DN*k2*n
// Lw = -kappa*i*ky*n   + alpha*(phi-n) - DOMEGA*k2*w ;  pack L = Ln + i*Lw
__global__ __launch_bounds__(256) void fft_lin_kernel(
    const float2* __restrict__ nh, const float2* __restrict__ wh,
    float2* __restrict__ dst, float dir, float scale)
{
    FFT_SHARED
    const int line = blockIdx.x;
    const size_t base = (size_t)line * NXY;
    const float ky = kfreq(line);
    for (int u = threadIdx.x; u < 2048; u += 256) {
        const float kx = kfreq(u);
        const float k2v = kx * kx + ky * ky;
        const float2 w = wh[base + u], nv = nh[base + u];
        const float inv = (k2v > K2MINf) ? (-1.0f / k2v) : 0.0f;
        const float2 ph = make_float2(inv * w.x, inv * w.y);
        const float2 dphidy = make_float2(-ky * ph.y, ky * ph.x);
        const float2 dndy   = make_float2(-ky * nv.y, ky * nv.x);
        const float cplr = ph.x - nv.x, cpli = ph.y - nv.y;
        const float Lnr = -KAPPAf * dphidy.x + ALPHAf * cplr - DNf * k2v * nv.x;
        const float Lni = -KAPPAf * dphidy.y + ALPHAf * cpli - DNf * k2v * nv.y;
        const float Lwr = -KAPPAf * dndy.x + ALPHAf * cplr - DOMEGAf * k2v * w.x;
        const float Lwi = -KAPPAf * dndy.y + ALPHAf * cpli - DOMEGAf * k2v * w.y;
        Ar[FFT_DEST(u)] = Lnr - Lwi;
        Ai[FFT_DEST(u)] = Lni + Lwr;
    }
    __syncthreads();
    fft_line_core<ST_PLAIN>(Ar, Ai, Br, Bi, dir, scale, line,
                            dst + base, nullptr, nullptr);
}

// -------- fwd pass1 with both Poisson brackets fused:  J = jn + i*jw -------
__global__ __launch_bounds__(256) void fft_jac_kernel(
    const float2* __restrict__ gphi, const float2* __restrict__ gn,
    const float2* __restrict__ gw, float2* __restrict__ dst,
    float dir, float scale)
{
    FFT_SHARED
    const int line = blockIdx.x;
    const size_t base = (size_t)line * NXY;
    for (int u = threadIdx.x; u < 2048; u += 256) {
        const float2 dp = gphi[base + u];
        const float2 dn = gn[base + u];
        const float2 dw = gw[base + u];
        Ar[FFT_DEST(u)] = dp.x * dn.y - dp.y * dn.x;   // jn
        Ai[FFT_DEST(u)] = dp.x * dw.y - dp.y * dw.x;   // jw
    }
    __syncthreads();
    fft_line_core<ST_PLAIN>(Ar, Ai, Br, Bi, dir, scale, line,
                            dst + base, nullptr, nullptr);
}

// -------- fwd pass2 of J with 2/3-dealias mask fused at the store ----------
__global__ __launch_bounds__(256) void fft_mask_kernel(
    float2* __restrict__ data, float dir, float scale)
{
    FFT_SHARED
    const int line = blockIdx.x;
    float2* ln = data + (size_t)line * NXY;
    for (int u = threadIdx.x; u < 2048; u += 256) {
        const float2 v = ln[u];
        Ar[FFT_DEST(u)] = v.x; Ai[FFT_DEST(u)] = v.y;
    }
    __syncthreads();
    fft_line_core<ST_MASK>(Ar, Ai, Br, Bi, dir, scale, line, ln, nullptr, nullptr);
}

// -------- final inv pass2 with combine fused: writes d_out directly --------
__global__ __launch_bounds__(256) void fft_combine_kernel(
    const float2* __restrict__ dataJ, const float2* __restrict__ Lreal,
    float* __restrict__ out, float dir, float scale)
{
    FFT_SHARED
    const int line = blockIdx.x;
    const float2* ln = dataJ + (size_t)line * NXY;
    for (int u = threadIdx.x; u < 2048; u += 256) {
        const float2 v = ln[u];
        Ar[FFT_DEST(u)] = v.x; Ai[FFT_DEST(u)] = v.y;
    }
    __syncthreads();
    fft_line_core<ST_COMBINE>(Ar, Ai, Br, Bi, dir, scale, line,
                              nullptr, Lreal, out);
}

// ---------------------------------------------------------------------------
// 32x32 float2 tiled transpose, coalesced both sides.  The global->LDS stage
// uses CDNA5 async copies (global_load_async_to_lds_b64, ASYNCcnt) so tile
// data never stages through VGPRs.
// ---------------------------------------------------------------------------
__global__ __launch_bounds__(256) void transpose_kernel(
    const float2* __restrict__ src, float2* __restrict__ dst)
{
    __shared__ float2 tile[32][33];        // 66-dword row stride: conflict-free
    const int tx = threadIdx.x & 31;
    const int ty = threadIdx.x >> 5;       // 8 row-groups
    const int x0 = (blockIdx.x & 63) * 32; // src column tile
    const int y0 = (blockIdx.x >> 6) * 32; // src row tile

#if defined(__AMDGCN__) && USE_ASYNC_TRANSPOSE
    #pragma unroll
    for (int j = 0; j < 32; j += 8) {
        const float2* gp = src + (size_t)(y0 + ty + j) * NXY + x0 + tx;
        const unsigned loff = (unsigned)(unsigned long long)(&tile[ty + j][tx]);
        asm volatile("global_load_async_to_lds_b64 %0, %1, off"
                     :: "v"(loff), "v"(gp) : "memory");
    }
  #if __has_builtin(__builtin_amdgcn_s_wait_asynccnt)
    __builtin_amdgcn_s_wait_asynccnt(0);
  #else
    asm volatile("s_wait_asynccnt 0x0" ::: "memory");
  #endif
#else
    #pragma unroll
    for (int j = 0; j < 32; j += 8)
        tile[ty + j][tx] = src[(size_t)(y0 + ty + j) * NXY + x0 + tx];
#endif
    __syncthreads();

    #pragma unroll
    for (int j = 0; j < 32; j += 8)
        dst[(size_t)(x0 + ty + j) * NXY + y0 + tx] = tile[tx][ty + j];
}

// Split Z = fft2(n + i*omega) into n_hat, w_hat (conjugate symmetry; the
// formula is layout-transpose invariant).
__global__ __launch_bounds__(256) void separate_kernel(
    const float2* __restrict__ Z, float2* __restrict__ nh, float2* __restrict__ wh)
{
    const int p = blockIdx.x * 256 + threadIdx.x;
    const int x = p >> 11, y = p & 2047;
    const int x2 = (NXY - x) & 2047, y2 = (NXY - y) & 2047;
    const float2 zk = Z[p];
    const float2 zm = Z[x2 * NXY + y2];
    const float a = zk.x, b = zk.y, c = zm.x, d = zm.y;
    nh[p] = make_float2(0.5f * (a + c), 0.5f * (b - d));
    wh[p] = make_float2(0.5f * (b + d), 0.5f * (c - a));
}

// ---------------------------------------------------------------------------
extern "C" void kernel_launch(void* const* d_in, const int* in_sizes, int n_in,
                              void* d_out, int out_size, void* d_ws, size_t ws_size,
                              hipStream_t stream)
{
    (void)in_sizes; (void)n_in; (void)out_size; (void)ws_size;
    const float* n_p = (const float*)d_in[0];
    const float* w_p = (const float*)d_in[1];
    float* out = (float*)d_out;

    float2* P0 = (float2*)d_ws;
    float2* P1 = P0 + NN;
    float2* P2 = P1 + NN;
    float2* P3 = P2 + NN;
    float2* P4 = P3 + NN;                 // 5 x 32MB = 160MB, L2-resident

    const dim3 pb(256), pg(NN / 256);     // pointwise
    const dim3 fb(256), fg(NXY);          // FFT lines
    const dim3 tb(256), tg(64 * 64);      // transpose tiles
    const float FWD = -1.0f, INV = 1.0f;
    const float S1 = 1.0f, SI = 1.0f / 2048.0f;

    // ---- forward: Z = fft2(n + i*omega), spectral layout [ky][kx] ----
    fft_pack_kernel<<<fg, fb, 0, stream>>>(n_p, w_p, P0, FWD, S1); // [x][ky]
    transpose_kernel<<<tg, tb, 0, stream>>>(P0, P1);               // [ky][x]
    fft_pass_kernel<<<fg, fb, 0, stream>>>(P1, FWD, S1);           // [ky][kx]
    separate_kernel<<<pg, pb, 0, stream>>>(P1, P0, P2);            // nh=P0 wh=P2

    // ---- inv pass1 with spectral combos fused (all read nh/wh) ----
    fft_lin_kernel<<<fg, fb, 0, stream>>>(P0, P2, P3, INV, SI);       // lin'
    fft_gradpack_kernel<<<fg, fb, 0, stream>>>(P2, P1, 1, INV, SI);   // gphi'
    fft_gradpack_kernel<<<fg, fb, 0, stream>>>(P0, P0, 0, INV, SI);   // gn'
    fft_gradpack_kernel<<<fg, fb, 0, stream>>>(P2, P2, 0, INV, SI);   // gw'

    // ---- transpose + inv pass2 -> real fields in natural layout ----
    transpose_kernel<<<tg, tb, 0, stream>>>(P1, P4);
    fft_pass_kernel<<<fg, fb, 0, stream>>>(P4, INV, SI);              // gphi
    transpose_kernel<<<tg, tb, 0, stream>>>(P0, P1);
    fft_pass_kernel<<<fg, fb, 0, stream>>>(P1, INV, SI);              // gn
    transpose_kernel<<<tg, tb, 0, stream>>>(P2, P0);
    fft_pass_kernel<<<fg, fb, 0, stream>>>(P0, INV, SI);              // gw
    transpose_kernel<<<tg, tb, 0, stream>>>(P3, P2);
    fft_pass_kernel<<<fg, fb, 0, stream>>>(P2, INV, SI);              // lin

    // ---- brackets: fwd FFT (jac fused) -> dealias (fused) -> inv FFT ----
    fft_jac_kernel<<<fg, fb, 0, stream>>>(P4, P1, P0, P3, FWD, S1);   // J' [x][ky]
    transpose_kernel<<<tg, tb, 0, stream>>>(P3, P4);                  // [ky][x]
    fft_mask_kernel<<<fg, fb, 0, stream>>>(P4, FWD, S1);              // [ky][kx]*mask
    fft_pass_kernel<<<fg, fb, 0, stream>>>(P4, INV, SI);              // [ky][x]
    transpose_kernel<<<tg, tb, 0, stream>>>(P4, P3);                  // [x][ky]
    fft_combine_kernel<<<fg, fb, 0, stream>>>(P3, P2, out, INV, SI);  // d_out
}